// GATLayer_63711544869187
// MI455X (gfx1250) — compile-verified
//
#include <hip/hip_runtime.h>
#include <hip/hip_bf16.h>

#define DIM       128
#define HEADS     4
#define HEAD_DIM  32

typedef __attribute__((ext_vector_type(2))) float v2f;
typedef __attribute__((ext_vector_type(8))) float v8f;

// Monotone float <-> uint mapping so atomicMax(u32) implements float max (incl. negatives).
__device__ __forceinline__ unsigned flipf(float f) {
    unsigned u = __float_as_uint(f);
    return (u & 0x80000000u) ? ~u : (u | 0x80000000u);
}
__device__ __forceinline__ float unflipf(unsigned u) {
    return __uint_as_float((u & 0x80000000u) ? (u & 0x7FFFFFFFu) : ~u);
}

// ---------------------------------------------------------------------------
// Kernel 1: Wx = x @ W^T via V_WMMA_F32_16X16X4_F32.
// One wave per 16-node tile. 8 accumulators (16 cols each) cover all 128 out
// dims; K-loop of 32 steps of 4 -> 256 WMMAs per wave.
// A layout (16x4 f32): lane&15 = M row, lane>>4 selects K pair {0,1}/{2,3}.
// B layout (4x16 f32): lane&15 = N col, lane>>4 selects K pair. B[k,n]=W[n,k].
// D layout: VGPR j, lanes 0-15 -> M=j, lanes 16-31 -> M=j+8, col = lane&15.
// ---------------------------------------------------------------------------
__global__ __launch_bounds__(256)
void gat_gemm_wmma(const float* __restrict__ x, const float* __restrict__ W,
                   float* __restrict__ Wx, int nNodes, int nTiles) {
    int wave = (blockIdx.x * blockDim.x + threadIdx.x) >> 5;
    if (wave >= nTiles) return;
    int lane = threadIdx.x & 31;
    int rc   = lane & 15;     // M row (for A) and N col (for B/D)
    int hi   = lane >> 4;     // K-half selector
    int nodeBase = wave * 16;

    v8f acc[8] = {};

    int arow = nodeBase + rc;
    if (arow >= nNodes) arow = nNodes - 1;          // clamp (stores guarded)
    const float* aptr = x + (size_t)arow * DIM + hi * 2;

    #pragma unroll 8
    for (int k = 0; k < DIM; k += 4) {
        v2f a = *(const v2f*)(aptr + k);
        #pragma unroll
        for (int nt = 0; nt < 8; ++nt) {
            v2f b = *(const v2f*)(W + (size_t)(nt * 16 + rc) * DIM + k + hi * 2);
            acc[nt] = __builtin_amdgcn_wmma_f32_16x16x4_f32(
                false, a, false, b, (short)0, acc[nt], false, false);
        }
    }

    #pragma unroll
    for (int nt = 0; nt < 8; ++nt) {
        #pragma unroll
        for (int j = 0; j < 8; ++j) {
            int node = nodeBase + j + 8 * hi;
            if (node < nNodes)
                Wx[(size_t)node * DIM + nt * 16 + rc] = acc[nt][j];
        }
    }
}

// ---------------------------------------------------------------------------
// Kernel 2: per-node attention scalars s_src[n,h], s_dst[n,h].
// ---------------------------------------------------------------------------
__global__ __launch_bounds__(256)
void gat_scores(const float* __restrict__ Wx, const float* __restrict__ a_w,
                float* __restrict__ s_src, float* __restrict__ s_dst, int nNodes) {
    int t = blockIdx.x * blockDim.x + threadIdx.x;
    if (t >= nNodes * HEADS) return;
    int h = t & (HEADS - 1);
    int n = t >> 2;
    const float* p = Wx + (size_t)n * DIM + h * HEAD_DIM;
    float ss = 0.f, sd = 0.f;
    #pragma unroll
    for (int d = 0; d < HEAD_DIM; ++d) {
        float v = p[d];
        ss += v * a_w[d];
        sd += v * a_w[HEAD_DIM + d];
    }
    s_src[t] = ss;
    s_dst[t] = sd;
}

// ---------------------------------------------------------------------------
// Kernel 3: segment max of leaky_relu(e) per (dst, head) via flipped atomicMax.
// ---------------------------------------------------------------------------
__global__ __launch_bounds__(256)
void gat_edge_max(const int* __restrict__ ei, const float* __restrict__ s_src,
                  const float* __restrict__ s_dst, unsigned* __restrict__ emax,
                  int nE) {
    int t = blockIdx.x * blockDim.x + threadIdx.x;
    if (t >= nE * HEADS) return;
    int h = t & (HEADS - 1);
    int e = t >> 2;
    int src = ei[e];
    int dst = ei[nE + e];
    float v = s_src[src * HEADS + h] + s_dst[dst * HEADS + h];
    v = v > 0.f ? v : 0.2f * v;
    atomicMax(&emax[dst * HEADS + h], flipf(v));
}

// ---------------------------------------------------------------------------
// Kernel 4: one wave per (edge, head). Compute e_exp, accumulate denom and
// e_exp * Wx_src into out (division by denom deferred to finalize).
// ---------------------------------------------------------------------------
__global__ __launch_bounds__(256)
void gat_edge_accum(const int* __restrict__ ei, const float* __restrict__ s_src,
                    const float* __restrict__ s_dst, const unsigned* __restrict__ emax,
                    const float* __restrict__ Wx, float* __restrict__ denom,
                    float* __restrict__ out, int nE) {
    int gw = (int)((blockIdx.x * (size_t)blockDim.x + threadIdx.x) >> 5);
    if (gw >= nE * HEADS) return;
    int lane = threadIdx.x & 31;
    int h = gw & (HEADS - 1);
    int e = gw >> 2;
    int src = ei[e];
    int dst = ei[nE + e];
    float v = s_src[src * HEADS + h] + s_dst[dst * HEADS + h];
    v = v > 0.f ? v : 0.2f * v;
    float em = unflipf(emax[dst * HEADS + h]);
    float ee = expf(v - em);
    if (lane == 0) atomicAdd(&denom[dst * HEADS + h], ee);
    atomicAdd(&out[(size_t)dst * DIM + h * HEAD_DIM + lane],
              ee * Wx[(size_t)src * DIM + h * HEAD_DIM + lane]);
}

// ---------------------------------------------------------------------------
// Kernel 5: out = elu(acc / (denom + 1e-8)), in place on d_out.
// ---------------------------------------------------------------------------
__global__ __launch_bounds__(256)
void gat_finalize(float* __restrict__ out, const float* __restrict__ denom,
                  int nNodes) {
    int t = blockIdx.x * blockDim.x + threadIdx.x;
    if (t >= nNodes * DIM) return;
    int n = t >> 7;
    int d = t & (DIM - 1);
    int h = d >> 5;
    float v = out[t] / (denom[n * HEADS + h] + 1e-8f);
    out[t] = v > 0.f ? v : expm1f(v);
}

extern "C" void kernel_launch(void* const* d_in, const int* in_sizes, int n_in,
                              void* d_out, int out_size, void* d_ws, size_t ws_size,
                              hipStream_t stream) {
    const float* x   = (const float*)d_in[0];
    const int*   ei  = (const int*)d_in[1];
    const float* W   = (const float*)d_in[2];
    const float* a_w = (const float*)d_in[3];
    int N = in_sizes[0] / DIM;
    int E = in_sizes[1] / 2;
    float* out = (float*)d_out;

    // Workspace layout: Wx | s_src | s_dst | denom | emax  (~29 MB for N=50000)
    float*    Wx    = (float*)d_ws;
    float*    s_src = Wx + (size_t)N * DIM;
    float*    s_dst = s_src + (size_t)N * HEADS;
    float*    denom = s_dst + (size_t)N * HEADS;
    unsigned* emax  = (unsigned*)(denom + (size_t)N * HEADS);

    hipMemsetAsync(out,   0, (size_t)N * DIM * sizeof(float), stream);
    hipMemsetAsync(denom, 0, (size_t)N * HEADS * sizeof(float), stream);
    hipMemsetAsync(emax,  0, (size_t)N * HEADS * sizeof(unsigned), stream);

    dim3 blk(256);
    int nTiles = (N + 15) / 16;
    gat_gemm_wmma<<<(nTiles + 7) / 8, blk, 0, stream>>>(x, W, Wx, N, nTiles);
    gat_scores  <<<(N * HEADS + 255) / 256, blk, 0, stream>>>(Wx, a_w, s_src, s_dst, N);
    gat_edge_max<<<((size_t)E * HEADS + 255) / 256, blk, 0, stream>>>(ei, s_src, s_dst, emax, E);
    gat_edge_accum<<<((size_t)E * HEADS * 32 + 255) / 256, blk, 0, stream>>>(
        ei, s_src, s_dst, emax, Wx, denom, out, E);
    gat_finalize<<<((size_t)N * DIM + 255) / 256, blk, 0, stream>>>(out, denom, N);
}